// GraphConvolution_1116691497131
// MI455X (gfx1250) — compile-verified
//
#include <hip/hip_runtime.h>
#include <stdint.h>

// ---------------------------------------------------------------------------
// GraphConvolution for MI455X (gfx1250, wave32):
//   1) support = x @ W          -> f32 WMMA (v_wmma_f32_16x16x4_f32),
//                                  A-tile staged via global_load_async_to_lds
//   2) out = 0                  -> vector zero fill
//   3) out[r] += v * support[c] -> L2-resident gather + global_atomic_add_f32
// ---------------------------------------------------------------------------

typedef float v2f __attribute__((ext_vector_type(2)));
typedef float v8f __attribute__((ext_vector_type(8)));

#define D_FEAT 256          // D_IN == D_OUT == 256
#define TILE_ROWS 32        // rows per block (2 M-tiles per wave)
#define A_STRIDE 260        // LDS row pad: bank step 4 per row -> conflict-free

// ---------------------------------------------------------------------------
// Dense GEMM: support[n, 256] = x[n, 256] @ W[256, 256]
// Block: 256 threads (8 waves). Block slab = 32 rows x 256 cols.
// Wave w: 32x32 output = 2 M-tiles x 2 N-tiles (cols [32w, 32w+32)).
// Per K-step: 2 A fragments + 2 B fragments -> 4 WMMAs (A and B each reused 2x).
// ---------------------------------------------------------------------------
__global__ __launch_bounds__(256) void gcn_gemm_wmma(
    const float* __restrict__ x, const float* __restrict__ w,
    float* __restrict__ support, int nRows) {
  __shared__ float aTile[TILE_ROWS * A_STRIDE];

  const int tid  = threadIdx.x;
  const int wave = tid >> 5;
  const int lane = tid & 31;
  const int half = lane >> 4;   // 0: lanes 0-15, 1: lanes 16-31
  const int lm   = lane & 15;
  const int rowBase  = blockIdx.x * TILE_ROWS;
  const bool fullTile = (rowBase + TILE_ROWS) <= nRows;

  if (fullTile) {
    // Async DMA of the 32x256 x-tile straight into LDS (ASYNCcnt-tracked,
    // no VGPR round-trip). 2048 x 16B chunks / 256 threads = 8 issues/thread.
#pragma unroll
    for (int i = 0; i < (TILE_ROWS * D_FEAT / 4) / 256; ++i) {
      const int idx = i * 256 + tid;
      const int r   = idx >> 6;          // row 0..31
      const int c4  = (idx & 63) << 2;   // col 0..252
      const unsigned ldsOff = (unsigned)(uintptr_t)&aTile[r * A_STRIDE + c4];
      const float* gp = x + (size_t)(rowBase + r) * D_FEAT + c4;
      asm volatile("global_load_async_to_lds_b128 %0, %1, off"
                   :: "v"(ldsOff), "v"(gp)
                   : "memory");
    }
    asm volatile("s_wait_asynccnt 0x0" ::: "memory");
  } else {
    // Ragged tail (never hit for N=100000): zero-padded synchronous staging.
    for (int i = tid; i < TILE_ROWS * (D_FEAT / 4); i += 256) {
      const int r  = i >> 6;
      const int c4 = (i & 63) << 2;
      float4 v = make_float4(0.f, 0.f, 0.f, 0.f);
      if (rowBase + r < nRows)
        v = *(const float4*)(x + (size_t)(rowBase + r) * D_FEAT + c4);
      float* dst = &aTile[r * A_STRIDE + c4];
      dst[0] = v.x; dst[1] = v.y; dst[2] = v.z; dst[3] = v.w;
    }
  }
  __syncthreads();

  const int colBase = wave << 5;    // 32 columns per wave
  v8f acc00 = {}, acc01 = {}, acc10 = {}, acc11 = {};
  const float* aRow0 = &aTile[lm * A_STRIDE];          // M-tile 0: rows 0..15
  const float* aRow1 = &aTile[(16 + lm) * A_STRIDE];   // M-tile 1: rows 16..31

#pragma unroll 4
  for (int k0 = 0; k0 < D_FEAT; k0 += 4) {
    const int kk = k0 + (half << 1);
    // A fragments (16x4 f32): lane holds A[M][kk], A[M][kk+1]
    v2f a0, a1;
    a0.x = aRow0[kk];
    a0.y = aRow0[kk + 1];
    a1.x = aRow1[kk];
    a1.y = aRow1[kk + 1];
    // B fragments (4x16 f32): lane holds W[kk][N], W[kk+1][N]
    const float* w0 = w + (size_t)kk * D_FEAT;
    v2f b0, b1;
    b0.x = w0[colBase + lm];
    b0.y = w0[D_FEAT + colBase + lm];
    b1.x = w0[colBase + 16 + lm];
    b1.y = w0[D_FEAT + colBase + 16 + lm];
    // 4 independent accumulation chains hide WMMA latency.
    acc00 = __builtin_amdgcn_wmma_f32_16x16x4_f32(
        false, a0, false, b0, (short)0, acc00, false, false);
    acc01 = __builtin_amdgcn_wmma_f32_16x16x4_f32(
        false, a0, false, b1, (short)0, acc01, false, false);
    acc10 = __builtin_amdgcn_wmma_f32_16x16x4_f32(
        false, a1, false, b0, (short)0, acc10, false, false);
    acc11 = __builtin_amdgcn_wmma_f32_16x16x4_f32(
        false, a1, false, b1, (short)0, acc11, false, false);
  }

  // C/D layout: VGPR j -> M=j (lanes 0-15) / M=j+8 (lanes 16-31), N = lm.
  if (fullTile) {
#pragma unroll
    for (int j = 0; j < 8; ++j) {
      const int m = j + (half << 3);
      float* o0 = support + (size_t)(rowBase + m) * D_FEAT + colBase;
      float* o1 = support + (size_t)(rowBase + 16 + m) * D_FEAT + colBase;
      o0[lm]      = acc00[j];
      o0[16 + lm] = acc01[j];
      o1[lm]      = acc10[j];
      o1[16 + lm] = acc11[j];
    }
  } else {
#pragma unroll
    for (int j = 0; j < 8; ++j) {
      const int m = j + (half << 3);
      const int r0 = rowBase + m;
      const int r1 = rowBase + 16 + m;
      if (r0 < nRows) {
        support[(size_t)r0 * D_FEAT + colBase + lm]      = acc00[j];
        support[(size_t)r0 * D_FEAT + colBase + 16 + lm] = acc01[j];
      }
      if (r1 < nRows) {
        support[(size_t)r1 * D_FEAT + colBase + lm]      = acc10[j];
        support[(size_t)r1 * D_FEAT + colBase + 16 + lm] = acc11[j];
      }
    }
  }
}

// ---------------------------------------------------------------------------
// Zero-fill of d_out (harness poisons it with 0xAA before timing).
// ---------------------------------------------------------------------------
__global__ __launch_bounds__(256) void zero_f32(float* __restrict__ p, int n) {
  const size_t i4 = ((size_t)blockIdx.x * 256 + threadIdx.x) * 4;
  if (i4 + 3 < (size_t)n) {
    *(float4*)(p + i4) = make_float4(0.f, 0.f, 0.f, 0.f);
  } else {
    for (size_t j = i4; j < (size_t)n; ++j) p[j] = 0.f;
  }
}

// ---------------------------------------------------------------------------
// SpMM scatter: for each edge (r, c, v): out[r, :] += v * support[c, :]
// One edge per 64 lanes; each lane owns 4 contiguous columns (128-bit gather).
// Edge scalars are wave-uniform -> single merged request per wave.
// support (102 MB) fits in the 192 MB L2 -> gathers and atomics resolve in L2.
// ---------------------------------------------------------------------------
__global__ __launch_bounds__(256) void gcn_spmm(
    const float* __restrict__ support, const int* __restrict__ erow,
    const int* __restrict__ ecol, const float* __restrict__ eval,
    float* __restrict__ out, int nEdges) {
  const long long e = (long long)blockIdx.x * 4 + (threadIdx.x >> 6);
  if (e >= nEdges) return;

  const int   r = erow[e];
  const int   c = ecol[e];
  const float v = eval[e];
  const int col = (threadIdx.x & 63) << 2;

  const float4 s = *(const float4*)(support + (size_t)c * D_FEAT + col);
  float* o = out + (size_t)r * D_FEAT + col;
  atomicAdd(o + 0, v * s.x);
  atomicAdd(o + 1, v * s.y);
  atomicAdd(o + 2, v * s.z);
  atomicAdd(o + 3, v * s.w);
}

// ---------------------------------------------------------------------------
// Launch: inputs in setup_inputs() order:
//   d_in[0] x[100000*256] f32, d_in[1] weight[256*256] f32,
//   d_in[2] edge_row[E] i32,   d_in[3] edge_col[E] i32, d_in[4] edge_val[E] f32
// d_out: out[100000*256] f32.  d_ws: support (102.4 MB).
// ---------------------------------------------------------------------------
extern "C" void kernel_launch(void* const* d_in, const int* in_sizes, int n_in,
                              void* d_out, int out_size, void* d_ws, size_t ws_size,
                              hipStream_t stream) {
  const float* x    = (const float*)d_in[0];
  const float* w    = (const float*)d_in[1];
  const int*   erow = (const int*)d_in[2];
  const int*   ecol = (const int*)d_in[3];
  const float* ev   = (const float*)d_in[4];
  float* out     = (float*)d_out;
  float* support = (float*)d_ws;

  const int nNodes = in_sizes[0] / D_FEAT;
  const int nEdges = in_sizes[2];

  // 1) out = 0
  {
    const int n4 = (out_size + 3) / 4;
    zero_f32<<<(n4 + 255) / 256, 256, 0, stream>>>(out, out_size);
  }
  // 2) support = x @ W  (WMMA f32, async-to-LDS A staging)
  gcn_gemm_wmma<<<(nNodes + TILE_ROWS - 1) / TILE_ROWS, 256, 0, stream>>>(
      x, w, support, nNodes);
  // 3) scatter-add over edges
  gcn_spmm<<<(nEdges + 3) / 4, 256, 0, stream>>>(support, erow, ecol, ev, out, nEdges);
}